// PredictiveCoding_90683939487861
// MI455X (gfx1250) — compile-verified
//
#include <hip/hip_runtime.h>
#include <math.h>

#define NDIM 8192
#define MDIM 8192
#define DDIM 256
#define MTILES (MDIM / 16)   // 512

#if defined(__has_builtin) && __has_builtin(__builtin_amdgcn_sched_barrier)
#define SCHED_FENCE() __builtin_amdgcn_sched_barrier(0)
#else
#define SCHED_FENCE()
#endif

typedef __attribute__((ext_vector_type(2))) float v2f;
typedef __attribute__((ext_vector_type(4))) float v4f;
typedef __attribute__((ext_vector_type(8))) float v8f;

// ---------------------------------------------------------------------------
// Fragment-pair layout (used for both the f_mu B operand in global memory and
// the x A operand staged in LDS):
// For a 16-column tile, fragment-pair p (0..31), lane (0..31):
//   v4f frag[p*32 + lane] =
//     { v(k=8p+2*hi,   col), v(k=8p+2*hi+1, col),      // K-fragment kf=2p
//       v(k=8p+4+2*hi, col), v(k=8p+4+2*hi+1, col) }   // K-fragment kf=2p+1
// where hi = lane>>4, col = (lane&15).  Matches the V_WMMA_F32_16X16X4_F32
// A/B operand layouts (A: lane<16 -> K={0,1}, lane>=16 -> K={2,3}).
// ---------------------------------------------------------------------------

// ---------------------------------------------------------------------------
// Phase 1: f[m][j] = tanh( sum_k mu[m][k] * W[j][k] + b[j] ), written straight
// into the WMMA B-fragment layout above (m -> column, j -> K).
// ---------------------------------------------------------------------------
__global__ __launch_bounds__(256) void fmu_kernel(const float* __restrict__ mu,
                                                  const float* __restrict__ W,
                                                  const float* __restrict__ bias,
                                                  float* __restrict__ frag)
{
    __shared__ float sMu[32][33];
    __shared__ float sW[32][33];
    const int tx = threadIdx.x & 31;   // m within tile
    const int ty = threadIdx.x >> 5;   // 0..7
    const int m0 = blockIdx.x * 32;
    const int j0 = blockIdx.y * 32;

    float acc[4] = {0.f, 0.f, 0.f, 0.f};   // j = j0 + ty + 8*i, m = m0 + tx
    for (int k0 = 0; k0 < DDIM; k0 += 32) {
        #pragma unroll
        for (int i = 0; i < 4; ++i) {
            const int r = ty + 8 * i;
            sMu[r][tx] = mu[(size_t)(m0 + r) * DDIM + k0 + tx];
            sW[r][tx]  = W [(size_t)(j0 + r) * DDIM + k0 + tx];
        }
        __syncthreads();
        #pragma unroll
        for (int kk = 0; kk < 32; ++kk) {
            const float mv = sMu[tx][kk];
            #pragma unroll
            for (int i = 0; i < 4; ++i)
                acc[i] = fmaf(mv, sW[ty + 8 * i][kk], acc[i]);
        }
        __syncthreads();
    }
    const int m  = m0 + tx;
    const int mt = m >> 4;
    const int n  = m & 15;
    #pragma unroll
    for (int i = 0; i < 4; ++i) {
        const int jj  = j0 + ty + 8 * i;
        const float f = tanhf(acc[i] + bias[jj]);
        const int kf  = jj >> 2;
        const int hi2 = (jj >> 1) & 1;
        const int sub = jj & 1;
        const int p   = kf >> 1;
        const int q   = kf & 1;
        const size_t idx =
            ((((size_t)mt * 32 + p) * 32) + hi2 * 16 + n) * 4 + q * 2 + sub;
        frag[idx] = f;
    }
}

// ---------------------------------------------------------------------------
// Phase 1b: fsq[col] = ||f_mu[col]||^2 from the fragment buffer.
// ---------------------------------------------------------------------------
__global__ __launch_bounds__(32) void fsq_kernel(const float* __restrict__ frag,
                                                 float* __restrict__ fsq)
{
    const int mt   = blockIdx.x;        // 0..511
    const int lane = threadIdx.x;       // 0..31
    const v4f* fp = reinterpret_cast<const v4f*>(frag) + (size_t)mt * 1024 + lane;
    float s = 0.f;
    #pragma unroll 8
    for (int p = 0; p < 32; ++p) {
        const v4f v = fp[p * 32];
        s = fmaf(v.x, v.x, s); s = fmaf(v.y, v.y, s);
        s = fmaf(v.z, v.z, s); s = fmaf(v.w, v.w, s);
    }
    s += __shfl_xor(s, 16, 32);         // combine K-halves per column
    if (lane < 16) fsq[mt * 16 + lane] = s;
}

// ---------------------------------------------------------------------------
// Phase 2: fused cross-GEMM (V_WMMA_F32_16X16X4_F32) + online logsumexp.
// Block = 256 threads = 8 waves = 2 row-tiles (16 rows) x 4 column groups.
// A operand (x rows) staged ONCE in LDS and re-read per tile (ds_load_b128,
// double-buffered; base index laundered per iteration so LICM cannot hoist
// the whole operand back into 128 VGPRs).
// B operand streamed from L2 as coalesced b128 loads through a 3-buffer
// rotation at prefetch distance 2; a sched_barrier after each prefetch block
// pins the loads ~8 deep in flight instead of letting the scheduler sink
// them to just-in-time.  Two independent accumulator chains.
// ---------------------------------------------------------------------------
__global__ __launch_bounds__(256) void score_lse_kernel(const float* __restrict__ x,
                                                        const float* __restrict__ frag,
                                                        const float* __restrict__ fsq,
                                                        float* __restrict__ partial)
{
    const int lane  = threadIdx.x & 31;
    const int wave  = threadIdx.x >> 5;   // 0..7
    const int rtile = wave >> 2;          // 0,1
    const int cgrp  = wave & 3;           // 0..3
    const int rlo   = lane & 15;
    const int hi    = lane >> 4;
    const int row_base = blockIdx.x * 32 + rtile * 16;

    __shared__ v4f   sA[2][32][32];       // 32 KB: A fragment pairs per row-tile
    __shared__ float sXsq[32];
    __shared__ float sMx[8][16];
    __shared__ float sSm[8][16];

    // Stage this wave's 8 assigned A fragment-pairs into LDS.
    const float* xrow = x + (size_t)(row_base + rlo) * DDIM + 2 * hi;
    #pragma unroll
    for (int j = 0; j < 8; ++j) {
        const int p = cgrp * 8 + j;
        const v2f t0 = *reinterpret_cast<const v2f*>(xrow + 8 * p);
        const v2f t1 = *reinterpret_cast<const v2f*>(xrow + 8 * p + 4);
        v4f q;
        q.x = t0.x; q.y = t0.y; q.z = t1.x; q.w = t1.y;
        sA[rtile][p][lane] = q;
    }
    __syncthreads();

    const v4f* ap = &sA[rtile][0][lane];

    // ||x_row||^2 per row from the staged fragments (one wave per row-tile).
    if (cgrp == 0) {
        float xs = 0.f;
        #pragma unroll 8
        for (int p = 0; p < 32; ++p) {
            const v4f v = ap[p * 32];
            xs = fmaf(v.x, v.x, xs); xs = fmaf(v.y, v.y, xs);
            xs = fmaf(v.z, v.z, xs); xs = fmaf(v.w, v.w, xs);
        }
        xs += __shfl_xor(xs, 16, 32);
        if (lane < 16) sXsq[rtile * 16 + lane] = xs;
    }
    __syncthreads();

    float xr[8];
    #pragma unroll
    for (int i = 0; i < 8; ++i) xr[i] = sXsq[rtile * 16 + i + 8 * hi];

    float mx[8], sm[8];
    #pragma unroll
    for (int i = 0; i < 8; ++i) { mx[i] = -INFINITY; sm[i] = 0.f; }

    const v4f* fragv4 = reinterpret_cast<const v4f*>(frag);

    for (int mt = cgrp; mt < MTILES; mt += 4) {
        // Opaque zero (fresh each iteration) so the A ds_loads below are not
        // loop-invariant and cannot be hoisted into resident VGPRs.
        int zz = 0;
        asm volatile("" : "+s"(zz));
        const v4f* alp = ap + zz;

        const v4f* fp = fragv4 + (size_t)mt * 1024 + lane;

        v4f bbuf[3][4];                     // B: 3-buffer rotation, distance 2
        v4f abuf[2][4];                     // A: double buffer, distance 1
        #pragma unroll
        for (int i = 0; i < 4; ++i) {
            bbuf[0][i] = fp[i * 32];
            bbuf[1][i] = fp[(4 + i) * 32];
            abuf[0][i] = alp[i * 32];
        }

        v8f acc0 = {};
        v8f acc1 = {};
        #pragma unroll
        for (int ch = 0; ch < 8; ++ch) {
            if (ch < 6) {                   // B prefetch, 2 chunks ahead
                #pragma unroll
                for (int i = 0; i < 4; ++i)
                    bbuf[(ch + 2) % 3][i] = fp[((ch + 2) * 4 + i) * 32];
            }
            if (ch < 7) {                   // A prefetch, 1 chunk ahead (LDS)
                #pragma unroll
                for (int i = 0; i < 4; ++i)
                    abuf[(ch + 1) & 1][i] = alp[((ch + 1) * 4 + i) * 32];
            }
            // Do not let the scheduler sink the prefetches below the WMMAs.
            SCHED_FENCE();
            #pragma unroll
            for (int i = 0; i < 4; ++i) {
                const v4f cb = bbuf[ch % 3][i];
                const v4f ca = abuf[ch & 1][i];
                const v2f alo = __builtin_shufflevector(ca, ca, 0, 1);
                const v2f ahi = __builtin_shufflevector(ca, ca, 2, 3);
                const v2f blo = __builtin_shufflevector(cb, cb, 0, 1);
                const v2f bhi = __builtin_shufflevector(cb, cb, 2, 3);
                if (i & 1) {
                    acc1 = __builtin_amdgcn_wmma_f32_16x16x4_f32(
                        false, alo, false, blo, (short)0, acc1, false, false);
                    acc1 = __builtin_amdgcn_wmma_f32_16x16x4_f32(
                        false, ahi, false, bhi, (short)0, acc1, false, false);
                } else {
                    acc0 = __builtin_amdgcn_wmma_f32_16x16x4_f32(
                        false, alo, false, blo, (short)0, acc0, false, false);
                    acc0 = __builtin_amdgcn_wmma_f32_16x16x4_f32(
                        false, ahi, false, bhi, (short)0, acc0, false, false);
                }
            }
        }

        const float fs = fsq[mt * 16 + rlo];
        #pragma unroll
        for (int i = 0; i < 8; ++i) {
            const float v  = (acc0[i] + acc1[i]) - 0.5f * (xr[i] + fs);
            const float nm = fmaxf(mx[i], v);
            sm[i] = sm[i] * __expf(mx[i] - nm) + __expf(v - nm);
            mx[i] = nm;
        }
    }

    // Merge (mx, sm) across the 16 lanes of each row group.
    #pragma unroll
    for (int off = 1; off < 16; off <<= 1) {
        #pragma unroll
        for (int i = 0; i < 8; ++i) {
            const float omx = __shfl_xor(mx[i], off, 32);
            const float osm = __shfl_xor(sm[i], off, 32);
            const float nm  = fmaxf(mx[i], omx);
            sm[i] = sm[i] * __expf(mx[i] - nm) + osm * __expf(omx - nm);
            mx[i] = nm;
        }
    }
    if (rlo == 0) {                 // lane 0 -> rows 0-7, lane 16 -> rows 8-15
        #pragma unroll
        for (int i = 0; i < 8; ++i) {
            sMx[wave][i + 8 * hi] = mx[i];
            sSm[wave][i + 8 * hi] = sm[i];
        }
    }
    __syncthreads();

    // Merge the 4 column-group waves per row-tile, take log, sum 32 row lse's.
    if (threadIdx.x < 32) {
        const int r = threadIdx.x & 15;
        const int g = threadIdx.x >> 4;         // 0: waves 0-3, 1: waves 4-7
        float M0 = sMx[4 * g][r], S0 = sSm[4 * g][r];
        #pragma unroll
        for (int w = 1; w < 4; ++w) {
            const float M1 = sMx[4 * g + w][r], S1 = sSm[4 * g + w][r];
            const float nm = fmaxf(M0, M1);
            S0 = S0 * __expf(M0 - nm) + S1 * __expf(M1 - nm);
            M0 = nm;
        }
        float lse = M0 + __logf(S0);
        #pragma unroll
        for (int off = 1; off < 32; off <<= 1)
            lse += __shfl_xor(lse, off, 32);
        if (threadIdx.x == 0) partial[blockIdx.x] = lse;
    }
}

// ---------------------------------------------------------------------------
// Phase 3: deterministic fixed-order reduction of 256 block partials.
// ---------------------------------------------------------------------------
__global__ __launch_bounds__(256) void final_reduce(const float* __restrict__ partial,
                                                    float* __restrict__ out)
{
    __shared__ float s[256];
    const int t = threadIdx.x;
    s[t] = partial[t];
    __syncthreads();
    for (int off = 128; off > 0; off >>= 1) {
        if (t < off) s[t] += s[t + off];
        __syncthreads();
    }
    if (t == 0) out[0] = -s[0];
}

// ---------------------------------------------------------------------------
extern "C" void kernel_launch(void* const* d_in, const int* in_sizes, int n_in,
                              void* d_out, int out_size, void* d_ws, size_t ws_size,
                              hipStream_t stream)
{
    const float* x  = (const float*)d_in[0];
    const float* mu = (const float*)d_in[1];
    const float* W  = (const float*)d_in[2];
    const float* b  = (const float*)d_in[3];
    float* out = (float*)d_out;

    char* ws = (char*)d_ws;
    float* frag    = (float*)ws;                                          // D*M f32 (8 MB)
    float* fsq     = (float*)(ws + (size_t)DDIM * MDIM * sizeof(float));  // M f32
    float* partial = fsq + MDIM;                                          // N/32 f32

    fmu_kernel<<<dim3(MDIM / 32, DDIM / 32), 256, 0, stream>>>(mu, W, b, frag);
    fsq_kernel<<<MTILES, 32, 0, stream>>>(frag, fsq);
    score_lse_kernel<<<NDIM / 32, 256, 0, stream>>>(x, frag, fsq, partial);
    final_reduce<<<1, 256, 0, stream>>>(partial, out);

    (void)in_sizes; (void)n_in; (void)out_size; (void)ws_size;
}